// Rank_IGR_Loss_13967233646995
// MI455X (gfx1250) — compile-verified
//
#include <hip/hip_runtime.h>
#include <hip/hip_bf16.h>

#define B_    8
#define N_    3125
#define SN    4096           // next pow2 >= N_, sort size
#define NSEG  (SN / 16)      // 256 segments of 16
#define GAM   3.0f

typedef __attribute__((ext_vector_type(2))) float v2f;
typedef __attribute__((ext_vector_type(8))) float v8f;

// Per-batch-item kernel: one workgroup (8 wave32) per item.
// Phase 1: IoU + prob, build (key, payload, idx) arrays, count positives K.
// Phase 2: LDS bitonic sort descending by (key, idx-ascending tie-break) --
//          matches jnp stable argsort(-key) semantics exactly.
// Phase 3: u = exp(-G*v), w = exp(+G*v) for ranks < K (else 0).
// Phase 4: S = sum_{a<b} u[a]*w[b] via fp32 WMMA triangular scan:
//          per 256-elem tile, D = T_strict x U (4 chained V_WMMA_F32_16X16X4_F32)
//          gives 16 exclusive prefix scans; dot with W -> intra-segment sums;
//          cross-segment handled with segment-sum running prefix.
__global__ __launch_bounds__(256) void rank_item_kernel(
    const float* __restrict__ cls, const int* __restrict__ lab,
    const float* __restrict__ pb, const float* __restrict__ tgt,
    float* __restrict__ ws)
{
    const int b   = blockIdx.x;
    const int tid = threadIdx.x;

    __shared__ float sKey[SN];
    __shared__ float sVal[SN];
    __shared__ int   sIdx[SN];
    __shared__ float sIntra[NSEG], sSU[NSEG], sSW[NSEG];
    __shared__ int   sK;

    const float tx1 = tgt[b * 4 + 0], ty1 = tgt[b * 4 + 1];
    const float tx2 = tgt[b * 4 + 2], ty2 = tgt[b * 4 + 3];
    const float ta  = (tx2 - tx1) * (ty2 - ty1);

    if (tid == 0) sK = 0;
    __syncthreads();
    int lp = 0;
    for (int n = tid; n < N_; n += 256) lp += (lab[b * N_ + n] > 0);
    atomicAdd(&sK, lp);
    __syncthreads();
    const int K = sK;

    for (int pass = 0; pass < 2; ++pass) {
        __syncthreads();
        // ---- build key/payload ----
        for (int n = tid; n < SN; n += 256) {
            float key = -3.0e38f, val = 0.f;
            if (n < N_ && lab[b * N_ + n] > 0) {
                float x1 = pb[(b * 4 + 0) * N_ + n];
                float y1 = pb[(b * 4 + 1) * N_ + n];
                float x2 = pb[(b * 4 + 2) * N_ + n];
                float y2 = pb[(b * 4 + 3) * N_ + n];
                float w0 = fmaxf(fminf(tx2, x2) - fmaxf(tx1, x1), 0.f);
                float h0 = fmaxf(fminf(ty2, y2) - fmaxf(ty1, y1), 0.f);
                float inter = w0 * h0;
                float area  = (x2 - x1) * (y2 - y1);
                float iou   = inter / (area + ta - inter);
                float prob  = expf(cls[(b * N_ + n) * 2 + 1]);
                key = (pass == 0) ? iou : prob;
                val = (pass == 0) ? prob : iou;
            }
            sKey[n] = key; sVal[n] = val; sIdx[n] = n;
        }
        __syncthreads();

        // ---- bitonic sort: descending key, ascending idx on ties ----
        for (int k = 2; k <= SN; k <<= 1) {
            for (int j = k >> 1; j > 0; j >>= 1) {
                for (int q = tid; q < SN / 2; q += 256) {
                    int i = ((q & ~(j - 1)) << 1) | (q & (j - 1));
                    int p = i | j;
                    bool dirDesc = ((i & k) == 0);
                    float ka = sKey[i], kb = sKey[p];
                    int   ia = sIdx[i], ib = sIdx[p];
                    bool g = (ka > kb) || (ka == kb && ia < ib);
                    if (g != dirDesc) {
                        sKey[i] = kb; sKey[p] = ka;
                        sIdx[i] = ib; sIdx[p] = ia;
                        float t0 = sVal[i]; sVal[i] = sVal[p]; sVal[p] = t0;
                    }
                }
                __syncthreads();
            }
        }

        // ---- exp transform; reuse sKey as u[], sIdx as w[] ----
        float* sU = sKey;
        float* sW = reinterpret_cast<float*>(sIdx);
        for (int n = tid; n < SN; n += 256) {
            float v = sVal[n];
            bool live = (n < K);
            sU[n] = live ? expf(-GAM * v) : 0.f;
            sW[n] = live ? expf( GAM * v) : 0.f;
        }
        __syncthreads();

        // ---- WMMA triangular scan (wave 0, EXEC all ones) ----
        if (tid < 32) {
            const int lane = tid;
            const int col  = lane & 15;   // N (and M for A) index
            const int half = lane >> 4;   // K-half selector per ISA f32 layout
            for (int t = 0; t < 16; ++t) {
                const int base = t * 256;
                v8f d = {};
#pragma unroll
                for (int c = 0; c < 4; ++c) {
                    // A 16x4 f32 layout: lane m=L%16; VGPR0 <- K=4c+2*half, VGPR1 <- +1
                    // B 4x16 f32 layout (mirrored): lane n=L%16, same K striping
                    int k0 = 4 * c + 2 * half;
                    v2f a, bb;
                    a.x = (k0     < col) ? 1.f : 0.f;   // strict lower triangular
                    a.y = (k0 + 1 < col) ? 1.f : 0.f;
                    bb.x = sU[base + 16 * col + k0];
                    bb.y = sU[base + 16 * col + k0 + 1];
                    d = __builtin_amdgcn_wmma_f32_16x16x4_f32(
                            false, a, false, bb, (short)0, d, false, false);
                }
                // D[m,n] = exclusive prefix of u within segment n; dot with w
                float part = 0.f;
#pragma unroll
                for (int j = 0; j < 8; ++j) {
                    int m = j + 8 * half;               // C/D layout: M = j + 8*(L/16)
                    part += d[j] * sW[base + 16 * col + m];
                }
                part += __shfl_xor(part, 16, 32);       // combine both M-halves
                if (half == 0) {
                    int seg = t * 16 + col;
                    float su = 0.f, sw = 0.f;
                    for (int m = 0; m < 16; ++m) {
                        su += sU[base + 16 * col + m];
                        sw += sW[base + 16 * col + m];
                    }
                    sIntra[seg] = part; sSU[seg] = su; sSW[seg] = sw;
                }
            }
        }
        __syncthreads();

        // ---- combine segments: S = sum intra + cross-segment prefix terms ----
        if (tid == 0) {
            float S = 0.f, run = 0.f;
            for (int s = 0; s < NSEG; ++s) {
                S   += sIntra[s] + sSW[s] * run;
                run += sSU[s];
            }
            ws[pass * B_ + b] = S;
            if (pass == 0) ws[2 * B_ + b] = (float)K;
        }
        __syncthreads();
    }
}

__global__ void rank_final_kernel(const float* __restrict__ ws,
                                  float* __restrict__ out)
{
    if (threadIdx.x == 0 && blockIdx.x == 0) {
        float s1 = 0.f, s2 = 0.f, nv = 0.f;
        for (int b = 0; b < B_; ++b) {
            float K   = ws[2 * B_ + b];
            float cnt = 0.5f * K * (K - 1.f);        // exact in f32 (cnt < 2^24)
            float den = fmaxf(cnt, 1.f);
            float l1  = ws[b] / den;
            float l2  = ws[B_ + b] / den;
            if (cnt > 0.f) { s1 += l1; s2 += l2; nv += 1.f; }
        }
        nv = fmaxf(nv, 1.f);
        out[0] = s1 / nv;
        out[1] = s2 / nv;
    }
}

extern "C" void kernel_launch(void* const* d_in, const int* in_sizes, int n_in,
                              void* d_out, int out_size, void* d_ws, size_t ws_size,
                              hipStream_t stream) {
    (void)in_sizes; (void)n_in; (void)out_size; (void)ws_size;
    const float* cls = (const float*)d_in[0];
    const int*   lab = (const int*)d_in[1];
    const float* pb  = (const float*)d_in[2];
    const float* tgt = (const float*)d_in[3];
    float* ws  = (float*)d_ws;
    float* out = (float*)d_out;

    rank_item_kernel<<<dim3(B_), dim3(256), 0, stream>>>(cls, lab, pb, tgt, ws);
    rank_final_kernel<<<dim3(1), dim3(32), 0, stream>>>(ws, out);
}